// YoloLoss_29772713296456
// MI455X (gfx1250) — compile-verified
//
#include <hip/hip_runtime.h>

// ---------------------------------------------------------------------------
// YOLO loss (forward only) for MI455X / gfx1250.
// Memory-bound streaming reduction: ~174 MB worst-case input, ~0.08 GFLOP.
// Bandwidth floor @23.3 TB/s ~= 7.5us; mask-gated loads cut real traffic to
// ~70-90 MB. Block-level reduction uses V_WMMA_F32_16X16X4_F32 (ones * B + C)
// as a 64-value column-summing engine in the epilogue.
// ---------------------------------------------------------------------------

typedef __attribute__((ext_vector_type(2))) float v2f;
typedef __attribute__((ext_vector_type(8))) float v8f;

namespace {
constexpr int   kS      = 14;
constexpr float kInvS   = 1.0f / 14.0f;
constexpr int   kNC     = 20;
constexpr int   kStride = 30;   // B*5 + NC
constexpr int   kBlock  = 256;  // 8 waves (wave32)
}

__device__ __forceinline__ float iou_xyxy(float ax1, float ay1, float ax2, float ay2,
                                          float bx1, float by1, float bx2, float by2) {
  float lx = fmaxf(ax1, bx1), ly = fmaxf(ay1, by1);
  float rx = fminf(ax2, bx2), ry = fminf(ay2, by2);
  float iw = fmaxf(rx - lx, 0.0f), ih = fmaxf(ry - ly, 0.0f);
  float inter = iw * ih;
  float a1 = (ax2 - ax1) * (ay2 - ay1);
  float a2 = (bx2 - bx1) * (by2 - by1);
  return inter / (a1 + a2 - inter);
}

// Reduce 4 per-thread accumulators across a 256-thread block.
// Packing: lds[t*4+c] = acc_c(thread t). Viewed as 16 chunks of a 4x16 f32
// B-tile (B[k][n] = lds[chunk*64 + k*16 + n]); since 16 % 4 == 0, column n of
// every chunk carries only component n%4. ones(16x4) x B + C accumulates the
// column sums into every row of C. Lane layout for f32 B 4x16 (2 VGPRs):
//   b[0]: lanes 0-15 -> K0,N=lane ; lanes 16-31 -> K1,N=lane-16  == lds[base+lane]
//   b[1]: lanes 0-15 -> K2        ; lanes 16-31 -> K3            == lds[base+32+lane]
// Returns (wave 0 only): lane n holds the total of component n&3.
__device__ __forceinline__ float block_reduce4_wmma(const float acc[4], float* lds) {
  const int t = threadIdx.x;
#pragma unroll
  for (int c = 0; c < 4; ++c) lds[t * 4 + c] = acc[c];
  __syncthreads();

  float v = 0.0f;
  if (t < 32) {                    // whole wave 0 active -> EXEC all ones
    v2f a; a[0] = 1.0f; a[1] = 1.0f;   // ones 16x4 A-tile
    v8f c = {};
#pragma unroll
    for (int j = 0; j < 16; ++j) {
      v2f b;
      b[0] = lds[j * 64 + t];
      b[1] = lds[j * 64 + 32 + t];
      c = __builtin_amdgcn_wmma_f32_16x16x4_f32(
          /*neg_a=*/false, a, /*neg_b=*/false, b,
          /*c_mod=*/(short)0, c, /*reuse_a=*/false, /*reuse_b=*/false);
    }
    v = c[0];                       // row M=0: lane n (n<16) = colsum[n]
    v += __shfl_xor(v, 8, 32);      // fold columns {n, n+8}
    v += __shfl_xor(v, 4, 32);      // fold columns {n, n+4} -> comp n&3
  }
  return v;
}

__global__ __launch_bounds__(kBlock)
void yolo_partial_kernel(const float* __restrict__ pred,
                         const float* __restrict__ tbox,
                         const float* __restrict__ tcls,
                         const unsigned char* __restrict__ mask,
                         float* __restrict__ partial,
                         int ncells) {
  __shared__ float lds[kBlock * 4];
  float acc[4] = {0.0f, 0.0f, 0.0f, 0.0f};  // reg, contain, noobj, cls

  for (int cell = blockIdx.x * blockDim.x + threadIdx.x; cell < ncells;
       cell += gridDim.x * blockDim.x) {
    const float* p = pred + (size_t)cell * kStride;   // 8-byte aligned
    const bool m = mask[cell] != 0;

    // Two predicted boxes: [x y w h conf] x 2 -> 5 x float2 loads (b64).
    float pb[10];
#pragma unroll
    for (int i = 0; i < 5; ++i) {
      v2f d = *(const v2f*)(p + 2 * i);
      pb[2 * i]     = d[0];
      pb[2 * i + 1] = d[1];
    }

    if (!m) {
      // no-object loss: both confidences squared
      acc[2] += pb[4] * pb[4] + pb[9] * pb[9];
    } else {
      // ---- class loss (gated: only ~15% of cells read these 40 floats) ----
      float cls = 0.0f;
#pragma unroll
      for (int i = 0; i < kNC / 4; ++i) {
        float4 tc = ((const float4*)(tcls + (size_t)cell * kNC))[i];
        v2f p0 = *(const v2f*)(p + 10 + 4 * i);
        v2f p1 = *(const v2f*)(p + 12 + 4 * i);
        float d0 = p0[0] - tc.x, d1 = p0[1] - tc.y;
        float d2 = p1[0] - tc.z, d3 = p1[1] - tc.w;
        cls += d0 * d0 + d1 * d1 + d2 * d2 + d3 * d3;
      }
      acc[3] += cls;

      // ---- IOU: pick responsible box ----
      float4 tb = ((const float4*)tbox)[cell];
      float tcx = tb.x * kInvS, tcy = tb.y * kInvS;
      float tx1 = tcx - 0.5f * tb.z, ty1 = tcy - 0.5f * tb.w;
      float tx2 = tcx + 0.5f * tb.z, ty2 = tcy + 0.5f * tb.w;

      float iou01[2];
#pragma unroll
      for (int k = 0; k < 2; ++k) {
        const float* b = pb + 5 * k;
        float cx = b[0] * kInvS, cy = b[1] * kInvS;
        float x1 = cx - 0.5f * b[2], y1 = cy - 0.5f * b[3];
        float x2 = cx + 0.5f * b[2], y2 = cy + 0.5f * b[3];
        iou01[k] = iou_xyxy(x1, y1, x2, y2, tx1, ty1, tx2, ty2);
      }
      const int best = (iou01[1] > iou01[0]) ? 1 : 0;   // argmax, first-max ties
      const float best_iou = fmaxf(iou01[0], iou01[1]);
      const float* r = pb + 5 * best;

      // ---- regression + containment ----
      float dx = r[0] - tb.x, dy = r[1] - tb.y;
      float dw = sqrtf(r[2]) - sqrtf(tb.z);
      float dh = sqrtf(r[3]) - sqrtf(tb.w);
      acc[0] += dx * dx + dy * dy + dw * dw + dh * dh;

      float dc = r[4] - best_iou;
      acc[1] += dc * dc;
    }
  }

  float v = block_reduce4_wmma(acc, lds);
  if (threadIdx.x < 4) partial[(size_t)blockIdx.x * 4 + threadIdx.x] = v;
}

__global__ __launch_bounds__(kBlock)
void yolo_finalize_kernel(const float* __restrict__ partial, int nparts, int n,
                          float* __restrict__ out) {
  __shared__ float lds[kBlock * 4];
  float acc[4] = {0.0f, 0.0f, 0.0f, 0.0f};
  for (int i = threadIdx.x; i < nparts; i += blockDim.x) {
#pragma unroll
    for (int c = 0; c < 4; ++c) acc[c] += partial[(size_t)i * 4 + c];
  }

  float v = block_reduce4_wmma(acc, lds);
  if (threadIdx.x < 32) {
    float reg     = __shfl(v, 0, 32);
    float contain = __shfl(v, 1, 32);
    float noobj   = __shfl(v, 2, 32);
    float cls     = __shfl(v, 3, 32);
    if (threadIdx.x == 0) {
      float total = (5.0f * reg + 0.5f * noobj + contain + cls) / (float)n;
      out[0] = total;
      out[1] = reg;
      out[2] = contain;
      out[3] = noobj;
      out[4] = cls;
    }
  }
}

extern "C" void kernel_launch(void* const* d_in, const int* in_sizes, int n_in,
                              void* d_out, int out_size, void* d_ws, size_t ws_size,
                              hipStream_t stream) {
  const float* pred         = (const float*)d_in[0];
  const float* tbox         = (const float*)d_in[1];
  const float* tcls         = (const float*)d_in[2];
  const unsigned char* mask = (const unsigned char*)d_in[3];

  const int ncells = in_sizes[0] / kStride;          // N * S * S
  const int n      = ncells / (kS * kS);             // batch N

  int blocks = (ncells + kBlock - 1) / kBlock;
  if (blocks > 1536) blocks = 1536;                  // 1536*8 = 12288 waves
  if (blocks < 1) blocks = 1;

  float* partial = (float*)d_ws;                     // blocks * 4 floats

  yolo_partial_kernel<<<blocks, kBlock, 0, stream>>>(pred, tbox, tcls, mask,
                                                     partial, ncells);
  yolo_finalize_kernel<<<1, kBlock, 0, stream>>>(partial, blocks, n,
                                                 (float*)d_out);
}